// VariantCoeLinear1dPredict_20787641712791
// MI455X (gfx1250) — compile-verified
//
#include <hip/hip_runtime.h>

#define N_CELLS 10000
#define BLOCK   1024

typedef unsigned int v4u __attribute__((ext_vector_type(4)));
typedef int          v4i __attribute__((ext_vector_type(4)));
typedef int          v8i __attribute__((ext_vector_type(8)));

// Device-pass-only TDM path (host pass must not see amdgcn target builtins).
#if defined(__HIP_DEVICE_COMPILE__) && \
    __has_builtin(__builtin_amdgcn_tensor_store_from_lds) && \
    __has_builtin(__builtin_amdgcn_tensor_load_to_lds)
#define USE_TDM 1
#else
#define USE_TDM 0
#endif

// Fixed polynomial flux (Horner, FMA chain).
__device__ __forceinline__ float flux_poly(float u) {
    float r = fmaf(5.1464175544821254e-05f, u, -0.005876625881704264f);
    r = fmaf(r, u,  0.23570323139025814f);
    r = fmaf(r, u, -0.8676433255901695f);
    r = fmaf(r, u,  0.06594985868545625f);
    r = fmaf(r, u,  1.5733140772285443f);
    r = fmaf(r, u,  0.4623734302522332f);
    return r;
}

#if USE_TDM
// Build a 1-row (2-D, tensor_dim1 = tile_dim1 = 1) descriptor moving
// N_CELLS f32 between a contiguous LDS row and a contiguous global row.
// D# layout per CDNA5 ISA §8: group0 = count/lds_addr/global_addr/type,
// group1 = data_size/dims/strides, groups 2/3 unused (zero).
__device__ __forceinline__ void tdm_make_row_desc(unsigned lds_byte_off,
                                                  const float* gptr,
                                                  v4u& g0, v8i& g1) {
    unsigned long long ga = (unsigned long long)(const void*)gptr;
    g0.x = 1u;                                              // count=1 (valid), user mode
    g0.y = lds_byte_off;                                    // lds_addr (bytes)
    g0.z = (unsigned)(ga & 0xFFFFFFFFu);                    // global_addr[31:0]
    g0.w = (unsigned)((ga >> 32) & 0x01FFFFFFu)             // global_addr[56:32]
         | 0x80000000u;                                     // type=2 ("image")
    g1 = (v8i){
        0x00020000,                                         // data_size=2 (4 bytes)
        (int)((N_CELLS & 0xFFFF) << 16),                    // tensor_dim0 lo16 (bits[47:32]=0)
        (int)((N_CELLS >> 16) | (1u << 16)),                // tensor_dim0 hi16 | tensor_dim1=1 lo16
        (int)((N_CELLS & 0xFFFF) << 16),                    // tensor_dim1 hi16=0 | tile_dim0
        1,                                                  // tile_dim1=1 | tile_dim2=0
        N_CELLS,                                            // tensor_dim0_stride[31:0]
        0,                                                  // stride hi16 | tensor_dim1_stride lo16
        0
    };
}

__device__ __forceinline__ void tdm_store_row(unsigned lds_byte_off, const float* gptr) {
    v4u g0; v8i g1;
    tdm_make_row_desc(lds_byte_off, gptr, g0, g1);
    v4i gz = {0, 0, 0, 0};
#if __clang_major__ >= 23
    v8i z8 = {0, 0, 0, 0, 0, 0, 0, 0};
    __builtin_amdgcn_tensor_store_from_lds(g0, g1, gz, gz, z8, 0);
#else
    __builtin_amdgcn_tensor_store_from_lds(g0, g1, gz, gz, 0);
#endif
}

__device__ __forceinline__ void tdm_load_row(unsigned lds_byte_off, const float* gptr) {
    v4u g0; v8i g1;
    tdm_make_row_desc(lds_byte_off, gptr, g0, g1);
    v4i gz = {0, 0, 0, 0};
#if __clang_major__ >= 23
    v8i z8 = {0, 0, 0, 0, 0, 0, 0, 0};
    __builtin_amdgcn_tensor_load_to_lds(g0, g1, gz, gz, z8, 0);
#else
    __builtin_amdgcn_tensor_load_to_lds(g0, g1, gz, gz, 0);
#endif
}
#endif  // USE_TDM

__global__ void lax_friedrichs_kernel(const float* __restrict__ init,
                                      const int*   __restrict__ stepnum_p,
                                      float*       __restrict__ out,
                                      int B) {
    const int b    = blockIdx.x;
    const int tid  = threadIdx.x;
    const int nthr = blockDim.x;
    const int steps = *stepnum_p;

    extern __shared__ float smem[];
    float* buf0 = smem;
    float* buf1 = smem + N_CELLS;

    const float dtdx  = 0.46f;   // DT/DX
    const float halfM = 0.8f;    // 0.5 * M_VISC

    const float* src  = init + (size_t)b * N_CELLS;
    float*       out0 = out  + (size_t)b * N_CELLS;

#if USE_TDM
    const bool wave0 = (tid < 32);
    // Prologue: DMA init row into LDS, then DMA it back out as trajectory
    // step 0. The wait(0) between them guarantees buf0 is populated before
    // the store reads it — and before any wave passes the first loop barrier.
    if (wave0) {
        tdm_load_row((unsigned)(size_t)buf0, src);
        __builtin_amdgcn_s_wait_tensorcnt(0);
        tdm_store_row((unsigned)(size_t)buf0, out0);   // acts as the "step -1" store of buf0
    }
#else
    for (int j = tid; j < N_CELLS; j += nthr) {
        float v = src[j];
        buf0[j] = v;
        __builtin_nontemporal_store(v, out0 + j);
    }
#endif
    __syncthreads();

    float* cur = buf0;
    float* nxt = buf1;
    const size_t row_stride = (size_t)B * N_CELLS;

    for (int s = 0; s < steps - 1; ++s) {
#if USE_TDM
        // Retire the TDM store issued 2 iterations ago (it read from the
        // buffer we are about to overwrite). TDM ops are in-order per wave,
        // so <=1 outstanding is exact. Only wave 0 ever issues TDM.
        if (wave0) __builtin_amdgcn_s_wait_tensorcnt(1);
#endif
        __syncthreads();   // safe-to-write 'nxt' + previous 'nxt' visible as 'cur'

        for (int j = 1 + tid; j < N_CELLS - 1; j += nthr) {
            float um = cur[j - 1];
            float u0 = cur[j];
            float up = cur[j + 1];
            float fm = flux_poly(um);
            float f0 = flux_poly(u0);
            float fp = flux_poly(up);
            float fhr = fmaf(0.5f, f0 + fp, -halfM * (up - u0));
            float fhl = fmaf(0.5f, fm + f0, -halfM * (u0 - um));
            float v = fmaf(-dtdx, fhr - fhl, u0);
            nxt[j] = v;
            if (j == 1)           nxt[0]           = v;  // Neumann boundaries
            if (j == N_CELLS - 2) nxt[N_CELLS - 1] = v;
        }
        __syncthreads();   // 'nxt' complete in LDS

        float* orow = out + (size_t)(s + 1) * row_stride + (size_t)b * N_CELLS;
#if USE_TDM
        if (wave0) {
            // Flat LDS addresses carry the LDS byte offset in addr[31:0] (ISA §10.2).
            tdm_store_row((unsigned)(size_t)nxt, orow);
        }
#else
        for (int j = tid; j < N_CELLS; j += nthr)
            __builtin_nontemporal_store(nxt[j], orow + j);
#endif
        float* t = cur; cur = nxt; nxt = t;
    }

#if USE_TDM
    // Drain DMA before wave teardown (S_ENDPGM also implies wait-idle).
    if (wave0) __builtin_amdgcn_s_wait_tensorcnt(0);
#endif
}

extern "C" void kernel_launch(void* const* d_in, const int* in_sizes, int n_in,
                              void* d_out, int out_size, void* d_ws, size_t ws_size,
                              hipStream_t stream) {
    const float* init    = (const float*)d_in[0];
    const int*   stepnum = (const int*)d_in[1];
    float*       out     = (float*)d_out;

    const int B = in_sizes[0] / N_CELLS;              // 64 rows, one workgroup each
    const size_t shmem = 2 * N_CELLS * sizeof(float); // 80 KB double-buffered row

    lax_friedrichs_kernel<<<dim3(B), dim3(BLOCK), shmem, stream>>>(init, stepnum, out, B);
}